// MPS_4664334484110
// MI455X (gfx1250) — compile-verified
//
#include <hip/hip_runtime.h>
#include <math.h>

typedef __attribute__((ext_vector_type(2))) float v2f;
typedef __attribute__((ext_vector_type(8))) float v8f;

#define BATCH   8192
#define NSITES  256
#define NBULK   254
#define BD      64
#define EPSV    1e-30f
#define MSTR    66          // padded LDS stride (floats) for transposed matrix cols
#define ESTR    66          // padded LDS stride (floats) for env rows
#define LN2F    0.6931471805599453f

static __device__ inline v8f wmma4(v2f a, v2f b, v8f c) {
  // D(16x16,f32) = A(16x4,f32) * B(4x16,f32) + C
  return __builtin_amdgcn_wmma_f32_16x16x4_f32(false, a, false, b, (short)0, c,
                                               false, false);
}

__global__ __launch_bounds__(256)
void mps_logamp2_kernel(const int* __restrict__ cfg,
                        const float* __restrict__ left,
                        const float* __restrict__ bulk,
                        const float* __restrict__ right,
                        float* __restrict__ out)
{
  // dynamic LDS partition (~110 KB of the 320 KB WGP pool)
  extern __shared__ __align__(16) float smem[];
  float*    sMT  = smem;                        // [2][128 * MSTR] transposed matrices
  float*    sEnv = sMT + 2 * 128 * MSTR;        // [8][16 * ESTR] per-wave env tiles
  unsigned* sCfgW = (unsigned*)(sEnv + 8 * 16 * ESTR);  // [8 * NBULK] config masks
  int*      sK   = (int*)(sCfgW + 8 * NBULK);   // [8 * 16] power-of-2 log-scale accum

  const int tid      = threadIdx.x;
  const int w        = tid >> 5;
  const int lane     = tid & 31;
  const int half     = lane >> 4;
  const int l16      = lane & 15;
  const int wgRow0   = blockIdx.x * 128;
  const int tileRow0 = wgRow0 + w * 16;

  // ---- precompute per-wave 16-bit config masks for all bulk sites ----
  for (int p = tid; p < 8 * NBULK; p += 256) {
    const int pw = p / NBULK;
    const int s  = p - pw * NBULK;
    const int rb = wgRow0 + pw * 16;
    unsigned mm = 0;
#pragma unroll 4
    for (int i = 0; i < 16; ++i)
      mm |= (cfg[(rb + i) * NSITES + s + 1] ? 1u : 0u) << i;
    sCfgW[p] = mm;
  }

  // ---- init env from left boundary; rescale by power of two ----
  if (lane < 16) {
    const int r  = lane;
    const int v0 = cfg[(tileRow0 + r) * NSITES + 0];
    const float* lp = left + v0 * BD;
    float mx = 0.f;
    for (int j = 0; j < BD; ++j) mx = fmaxf(mx, fabsf(lp[j]));
    const float m = fmaxf(mx, EPSV);
    const int e = (int)(__float_as_uint(m) >> 23) - 127;       // floor(log2(m))
    const float sc = __uint_as_float((unsigned)(127 - e) << 23); // exact 2^-e
    for (int j = 0; j < BD; ++j)
      sEnv[(w * 16 + r) * ESTR + j] = lp[j] * sc;
    sK[w * 16 + r] = e;
  }

  // ---- stage site 0 (transposed) into LDS buffer 0 ----
  {
    const float4* src = (const float4*)bulk;
    float4 rg[8];
#pragma unroll
    for (int u = 0; u < 8; ++u) rg[u] = src[tid + u * 256];
#pragma unroll
    for (int u = 0; u < 8; ++u) {
      const int p4 = (tid + u * 256) * 4;   // flat float index
      const int i  = p4 >> 7;               // matrix row (K index)
      const int cc = p4 & 127;              // combined col v*64+j
      sMT[(cc + 0) * MSTR + i] = rg[u].x;
      sMT[(cc + 1) * MSTR + i] = rg[u].y;
      sMT[(cc + 2) * MSTR + i] = rg[u].z;
      sMT[(cc + 3) * MSTR + i] = rg[u].w;
    }
  }
  __syncthreads();

  // ---- chain over bulk sites (one barrier per site, double-buffered) ----
  for (int t = 0; t < NBULK; ++t) {
    // issue next site's global loads into registers (hidden behind compute)
    float4 rg[8];
    const bool more = (t + 1 < NBULK);
    if (more) {
      const float4* src = (const float4*)(bulk + (size_t)(t + 1) * (BD * 2 * BD));
#pragma unroll
      for (int u = 0; u < 8; ++u) rg[u] = src[tid + u * 256];
    }
    if (t + 2 < NBULK)   // warm L2 for the site after that (global_prefetch_b8)
      __builtin_prefetch((const char*)(bulk + (size_t)(t + 2) * (BD * 2 * BD)) + tid * 128, 0, 0);

    const float*   Mb  = sMT + (t & 1) * 128 * MSTR;
    const unsigned m16 = sCfgW[w * NBULK + t];
    const float* envRow = sEnv + w * 16 * ESTR;

    // A fragments: env(16x64), lane holds (row=l16, K=4kb+2*half,+1) -> ds_load_b64
    v2f a[16];
#pragma unroll
    for (int kb = 0; kb < 16; ++kb)
      a[kb] = *(const v2f*)(envRow + l16 * ESTR + 4 * kb + 2 * half);

    float envN[4][8];
    float amax[8];
#pragma unroll
    for (int r = 0; r < 8; ++r) amax[r] = 0.f;

#pragma unroll
    for (int n = 0; n < 4; ++n) {          // output N tiles of 16 columns
      v8f d0 = {}; v8f d1 = {};
#pragma unroll
      for (int kb = 0; kb < 16; ++kb) {    // K steps of 4
        const int ko = 4 * kb + 2 * half;  // this lane's K pair
        // transposed layout: (col, K..K+1) contiguous -> single ds_load_b64 each
        const v2f b0 = *(const v2f*)(Mb + (n * 16 + l16) * MSTR + ko);        // v=0
        const v2f b1 = *(const v2f*)(Mb + ((n + 4) * 16 + l16) * MSTR + ko);  // v=1
        d0 = wmma4(a[kb], b0, d0);
        d1 = wmma4(a[kb], b1, d1);
      }
#pragma unroll
      for (int r = 0; r < 8; ++r) {        // per-row branch select
        const int rr = r + half * 8;
        const float val = ((m16 >> rr) & 1u) ? d1[r] : d0[r];
        envN[n][r] = val;
        amax[r] = fmaxf(amax[r], fabsf(val));
      }
    }

    // row maxima across the 16 lanes of each half-wave
#pragma unroll
    for (int r = 0; r < 8; ++r) {
      float x = amax[r];
      x = fmaxf(x, __shfl_xor(x, 1, 32));
      x = fmaxf(x, __shfl_xor(x, 2, 32));
      x = fmaxf(x, __shfl_xor(x, 4, 32));
      x = fmaxf(x, __shfl_xor(x, 8, 32));
      amax[r] = fmaxf(x, EPSV);
    }

    // power-of-2 rescale: exact scaling, integer log accumulation (no logf/div)
    int   ex[8];
    float sc[8];
#pragma unroll
    for (int r = 0; r < 8; ++r) {
      ex[r] = (int)(__float_as_uint(amax[r]) >> 23) - 127;
      sc[r] = __uint_as_float((unsigned)(127 - ex[r]) << 23);
    }
    if (l16 == 0) {                        // lane 0 -> rows 0..7, lane 16 -> 8..15
#pragma unroll
      for (int r = 0; r < 8; ++r)
        sK[w * 16 + half * 8 + r] += ex[r];
    }

    // write rescaled env back row-major (D layout -> A layout via LDS)
#pragma unroll
    for (int n = 0; n < 4; ++n)
#pragma unroll
      for (int r = 0; r < 8; ++r)
        sEnv[w * 16 * ESTR + (r + half * 8) * ESTR + n * 16 + l16] =
            envN[n][r] * sc[r];

    // commit next site's matrices (transposed) into the other LDS buffer
    if (more) {
      float* Mn = sMT + ((t + 1) & 1) * 128 * MSTR;
#pragma unroll
      for (int u = 0; u < 8; ++u) {
        const int p4 = (tid + u * 256) * 4;
        const int i  = p4 >> 7;
        const int cc = p4 & 127;
        Mn[(cc + 0) * MSTR + i] = rg[u].x;
        Mn[(cc + 1) * MSTR + i] = rg[u].y;
        Mn[(cc + 2) * MSTR + i] = rg[u].z;
        Mn[(cc + 3) * MSTR + i] = rg[u].w;
      }
    }

    __syncthreads();
  }

  // ---- right boundary contraction + output ----
  if (lane < 16) {
    const int r  = lane;
    const int b  = tileRow0 + r;
    const int vL = cfg[b * NSITES + (NSITES - 1)];
    const float* er = &sEnv[(w * 16 + r) * ESTR];
    float psi = 0.f;
    for (int j = 0; j < BD; ++j)
      psi += er[j] * right[j * 2 + vL];
    const float mm = fmaxf(fabsf(psi), EPSV);
    const float ls = (float)sK[w * 16 + r] * LN2F + logf(mm);
    const float pr = psi / mm;
    const float a2 = fmaxf(pr * pr, EPSV);
    out[b] = logf(a2) + 2.0f * ls;
  }
}

extern "C" void kernel_launch(void* const* d_in, const int* in_sizes, int n_in,
                              void* d_out, int out_size, void* d_ws, size_t ws_size,
                              hipStream_t stream) {
  const int*   cfg   = (const int*)d_in[0];
  const float* left  = (const float*)d_in[1];
  const float* bulk  = (const float*)d_in[2];
  const float* right = (const float*)d_in[3];
  float* out = (float*)d_out;

  const size_t lds_bytes =
      (size_t)(2 * 128 * MSTR + 8 * 16 * ESTR) * sizeof(float) +
      (size_t)(8 * NBULK) * sizeof(unsigned) +
      (size_t)(8 * 16) * sizeof(int);

  dim3 grid(BATCH / 128);   // 128 batch rows per workgroup (8 waves x 16)
  dim3 block(256);
  hipLaunchKernelGGL(mps_logamp2_kernel, grid, block, lds_bytes, stream,
                     cfg, left, bulk, right, out);
}